// CRF_79551384256937
// MI455X (gfx1250) — compile-verified
//
#include <hip/hip_runtime.h>

// CRF forward scan as exp-space GEMM on CDNA5 WMMA (V_WMMA_F32_16X16X4_F32).
//
// Shapes: B=128, T=1024, NT=254, N=256 (start=254, end=255), NEG=-10000.
// Recurrence per (b):  alpha'[j] = u_t[j] + logsumexp_k(alpha[k] + trans[j,k])
//   => with E = exp(trans), p[k] = exp(alpha[k]-m):  alpha'[j] = m + log((E p)[j]) + u_t[j]
// Each workgroup owns 16 batch columns; E (256x256 f32, exp-space) lives in the
// 320KB WGP LDS; P (256x16) panel is rebuilt each step with running-max rescale.
// Next step's unary tile is double-buffered into LDS with async-to-LDS copies.

#define B_TOT    128
#define T_LEN    1024
#define NT       254
#define NTAG     256
#define START_I  254
#define END_I    255
#define NEGV     (-10000.0f)

#define BPW      16      // batch columns per workgroup (WMMA N tile)
#define NWAVES   16      // 512 threads, one 16-row M-tile per wave
#define ROWPAD   260     // padded row stride (floats): 260 % 64 == 4 -> no bank conflicts
#define UROW     260     // padded unary row stride in LDS
#define NCHUNK   (BPW * 127)   // 2032 b64 chunks per unary tile

typedef float v2f __attribute__((ext_vector_type(2)));
typedef float v8f __attribute__((ext_vector_type(8)));
typedef int   b64v __attribute__((vector_size(2 * sizeof(int))));  // async b64 payload type

#define AS1 __attribute__((address_space(1)))
#define AS3 __attribute__((address_space(3)))

#if defined(__has_builtin)
# if __has_builtin(__builtin_amdgcn_global_load_async_to_lds_b64)
#  define HAVE_ASYNC_LDS 1
# endif
#endif

static __device__ __forceinline__ void wait_async0() {
#if defined(__has_builtin) && __has_builtin(__builtin_amdgcn_s_wait_asynccnt)
    __builtin_amdgcn_s_wait_asynccnt(0);
#else
    asm volatile("s_wait_asynccnt 0x0" ::: "memory");
#endif
}

// Raw-hardware log/exp: inputs are guaranteed away from denormal/inf here
// (one P column is always exactly 1.0 and E = exp(trans) ~ 1, so acc in [0.9,256];
// exp args are <= 0). v_log_f32 / v_exp_f32 + one scale op each.
static __device__ __forceinline__ float fast_log(float x) {
#if defined(__has_builtin) && __has_builtin(__builtin_amdgcn_logf)
    return __builtin_amdgcn_logf(x) * 0.69314718055994531f;
#else
    return __logf(x);
#endif
}
static __device__ __forceinline__ float fast_exp(float x) {
#if defined(__has_builtin) && __has_builtin(__builtin_amdgcn_exp2f)
    return __builtin_amdgcn_exp2f(x * 1.44269504088896340f);
#else
    return __expf(x);
#endif
}

__global__ __launch_bounds__(512)
void crf_forward_kernel(const float* __restrict__ unary,
                        const long long* __restrict__ lengths,
                        const float* __restrict__ trans,
                        float* __restrict__ fwd_out)
{
    extern __shared__ float smem[];
    float* Elds = smem;                       // [256][260]  exp(trans), row-major padded
    float* pT   = Elds + NTAG * ROWPAD;       // [16][260]   P^T: pT[b][k] = exp(alpha_hat[k])
    float* red  = pT   + BPW  * ROWPAD;       // [16][16]    per-wave column partials
    float* mlds = red  + NWAVES * BPW;        // [16]        running max per column
    float* ulds = mlds + BPW;                 // [2][16][260] double-buffered unary tile

    const int tid  = threadIdx.x;
    const int wave = tid >> 5;
    const int lane = tid & 31;
    const int h    = lane >> 4;      // half-wave (K / row-group select)
    const int bcol = lane & 15;      // batch column within tile
    const int m0   = wave * 16;      // this wave's row-tile base
    const int b_base = blockIdx.x * BPW;
    const int bg   = b_base + bcol;

    // ---- init: E = exp(trans) resident in LDS; P = onehot(start); m = 0 ----
    for (int idx = tid; idx < NTAG * NTAG; idx += blockDim.x) {
        int j = idx >> 8, k = idx & 255;
        Elds[j * ROWPAD + k] = expf(trans[idx]);
    }
    for (int idx = tid; idx < BPW * ROWPAD; idx += blockDim.x) pT[idx] = 0.0f;
    if (tid < BPW) { pT[tid * ROWPAD + START_I] = 1.0f; mlds[tid] = 0.0f; }
    // padding columns of the unary tile hold NEG so row reads need no bounds select
    if (tid < 2 * BPW) {
        for (int c = NT; c < UROW; ++c) ulds[tid * UROW + c] = NEGV;
    }

#ifdef HAVE_ASYNC_LDS
    // per-thread chunk plan (step-invariant): 4 b64 chunks of the 16x254 tile
    const float* gsrc[4];   // global base without the t*NT term
    float*       ldst0[4];  // LDS dest in buffer 0
    float*       ldst1[4];  // LDS dest in buffer 1
    bool         cvalid[4];
    #pragma unroll
    for (int i = 0; i < 4; ++i) {
        int c = tid + 512 * i;
        cvalid[i] = (c < NCHUNK);
        int cc = cvalid[i] ? c : 0;
        int r = cc / 127, k2 = (cc % 127) * 2;
        gsrc[i]  = unary + (long)(b_base + r) * T_LEN * NT + k2;
        ldst0[i] = ulds + r * UROW + k2;
        ldst1[i] = ulds + (BPW * UROW) + r * UROW + k2;
    }
    // prime buffer 0 with the t=0 unary tile
    #pragma unroll
    for (int i = 0; i < 4; ++i) {
        if (cvalid[i])
            __builtin_amdgcn_global_load_async_to_lds_b64(
                (AS1 b64v*)gsrc[i], (AS3 b64v*)ldst0[i], 0, 0);
    }
    wait_async0();
#endif
    __syncthreads();

    const int   lenb  = (int)lengths[bg];
    const int   rowA  = m0 + (lane & 15);           // A-matrix M index (per ISA layout)
    const float* Arow = Elds + rowA * ROWPAD;
    const float* Brow = pT   + bcol * ROWPAD;
    const int   jbase = m0 + 8 * h;                 // first C/D row this lane owns

    for (int t = 0; t < T_LEN; ++t) {
#ifdef HAVE_ASYNC_LDS
        // kick off async fill of next step's tile into the other buffer
        if (t + 1 < T_LEN) {
            const long toff = (long)(t + 1) * NT;
            float* const* ld = ((t + 1) & 1) ? ldst1 : ldst0;
            #pragma unroll
            for (int i = 0; i < 4; ++i) {
                if (cvalid[i])
                    __builtin_amdgcn_global_load_async_to_lds_b64(
                        (AS1 b64v*)(gsrc[i] + toff), (AS3 b64v*)ld[i], 0, 0);
            }
        }
#endif
        float mb = mlds[bcol];

        // this step's unary values for rows jbase..jbase+7
        float uv[8];
#ifdef HAVE_ASYNC_LDS
        {
            // unconditional, 16B-aligned: padding columns carry NEG
            const float* urow = ulds + (t & 1) * (BPW * UROW) + bcol * UROW + jbase;
            #pragma unroll
            for (int v = 0; v < 8; ++v) uv[v] = urow[v];
        }
#else
        {
            const long ubase = ((long)bg * T_LEN + t) * NT;
            #pragma unroll
            for (int v = 0; v < 8; ++v) {
                int j = jbase + v;
                uv[v] = (j < NT) ? unary[ubase + j] : NEGV;
            }
            if (t + 1 < T_LEN)
                __builtin_prefetch(&unary[ubase + NT + jbase], 0, 0);
        }
#endif

        // D(16x16) = E_tile(16x256) * P(256x16), K in chunks of 4 (f32 WMMA)
        v8f acc = {};
        #pragma unroll 8
        for (int kc = 0; kc < 64; ++kc) {
            int k = kc * 4 + 2 * h;
            v2f a = *(const v2f*)(Arow + k);   // A: M=lane&15, K = 2h + vgpr
            v2f b = *(const v2f*)(Brow + k);   // B: N=lane&15, K = 2h + vgpr
            acc = __builtin_amdgcn_wmma_f32_16x16x4_f32(
                false, a, false, b, (short)0, acc, false, false);
        }

        // new alpha for this lane's 8 rows; per-column running max
        float na[8];
        float pm = -3.402823466e38f;
        #pragma unroll
        for (int v = 0; v < 8; ++v) {          // C/D: vgpr v -> row m0 + v + 8h
            na[v] = mb + fast_log(acc[v]) + uv[v];
            pm = fmaxf(pm, na[v]);
        }
        pm = fmaxf(pm, __shfl_xor(pm, 16, 32));        // merge the two half-wave row groups
        if (h == 0) red[wave * BPW + bcol] = pm;
        __syncthreads();

        float nm = -3.402823466e38f;
        #pragma unroll
        for (int w = 0; w < NWAVES; ++w) nm = fmaxf(nm, red[w * BPW + bcol]);

        if (t < lenb) {                                 // masked scan update
            float* prow = pT + bcol * ROWPAD + jbase;
            #pragma unroll
            for (int v = 0; v < 8; ++v) prow[v] = fast_exp(na[v] - nm);  // args <= 0
            if (wave == 0 && h == 0) mlds[bcol] = nm;
        }
#ifdef HAVE_ASYNC_LDS
        wait_async0();   // next-step tile (issued above) is now resident
#endif
        __syncthreads();
    }

    // ---- fwd_score[b] = m_b + log( sum_j p[j][b] * exp(trans[end, j]) ) ----
    {
        float mb = mlds[bcol];
        float s = 0.0f;
        #pragma unroll
        for (int v = 0; v < 8; ++v) {
            int j = jbase + v;
            s += pT[bcol * ROWPAD + j] * expf(trans[END_I * NTAG + j]);
        }
        s += __shfl_xor(s, 16, 32);
        if (h == 0) red[wave * BPW + bcol] = s;
        __syncthreads();
        if (wave == 0 && h == 0) {
            float tot = 0.0f;
            #pragma unroll
            for (int w = 0; w < NWAVES; ++w) tot += red[w * BPW + bcol];
            fwd_out[bg] = mb + logf(tot);
        }
    }
}

// Gold path score + final subtraction: trivial sequential gather per batch.
__global__ void crf_gold_kernel(const float* __restrict__ unary,
                                const long long* __restrict__ tags,
                                const long long* __restrict__ lengths,
                                const float* __restrict__ trans,
                                const float* __restrict__ fwd,
                                float* __restrict__ out)
{
    int b = blockIdx.x * blockDim.x + threadIdx.x;
    if (b >= B_TOT) return;
    int len  = (int)lengths[b];
    int prev = START_I;
    float gold = 0.0f;
    for (int i = 0; i < len; ++i) {
        int cur = (int)tags[(long)b * T_LEN + i];
        gold += trans[cur * NTAG + prev] + unary[((long)b * T_LEN + i) * NT + cur];
        prev = cur;
    }
    gold += trans[END_I * NTAG + prev];
    out[b] = fwd[b] - gold;
}

extern "C" void kernel_launch(void* const* d_in, const int* in_sizes, int n_in,
                              void* d_out, int out_size, void* d_ws, size_t ws_size,
                              hipStream_t stream)
{
    const float*      unary   = (const float*)d_in[0];
    const long long*  tags    = (const long long*)d_in[1];   // jnp.int64
    const long long*  lengths = (const long long*)d_in[2];   // jnp.int64
    const float*      trans   = (const float*)d_in[3];
    float*            out     = (float*)d_out;
    float*            fwd     = (float*)d_ws;                // 128 floats of scratch

    const size_t smem_bytes =
        (size_t)(NTAG * ROWPAD + BPW * ROWPAD + NWAVES * BPW + BPW
                 + 2 * BPW * UROW) * sizeof(float);
    // ~317 KB dynamic LDS: fits CDNA5's 320 KB per-workgroup allocation.

    hipLaunchKernelGGL(crf_forward_kernel, dim3(B_TOT / BPW), dim3(512),
                       smem_bytes, stream, unary, lengths, trans, fwd);
    hipLaunchKernelGGL(crf_gold_kernel, dim3(1), dim3(B_TOT),
                       0, stream, unary, tags, lengths, trans, fwd, out);
}